// TransConv_19112604467378
// MI455X (gfx1250) — compile-verified
//
#include <hip/hip_runtime.h>

// ---- problem constants (match reference) ----
constexpr int B_  = 64;
constexpr int M_  = 512;
constexpr int IN_ = 256;
constexpr int H_  = 4;
constexpr int D_  = 64;
constexpr int N_  = B_ * M_;     // 32768
constexpr int E_  = N_ * 16;     // 524288
constexpr int HD_ = H_ * D_;     // 256

// ---- workspace layout (float offsets) ----
constexpr size_t O_Q    = 0;
constexpr size_t O_K    = O_Q    + (size_t)N_ * HD_;        //  8388608
constexpr size_t O_V    = O_K    + (size_t)N_ * HD_;        // 16777216
constexpr size_t O_VH   = O_V    + (size_t)N_ * HD_;        // 25165824
constexpr size_t O_KV   = O_VH   + (size_t)N_ * D_;         // 27262976
constexpr size_t O_KSUM = O_KV   + (size_t)B_ * H_ * D_ * D_;
constexpr size_t O_VSUM = O_KSUM + (size_t)B_ * H_ * D_;
constexpr size_t O_DEG  = O_VSUM + (size_t)B_ * H_ * D_;
constexpr size_t O_SSQ  = O_DEG  + (size_t)N_;              // 2 floats right after deg

typedef __attribute__((ext_vector_type(2))) float v2f;
typedef __attribute__((ext_vector_type(8))) float v8f;

// ---------------------------------------------------------------------------
// zero-init deg (N) + sumsq (2) each call (harness does not re-poison)
__global__ void zero_kernel(float* __restrict__ p, int n) {
    int i = blockIdx.x * blockDim.x + threadIdx.x;
    if (i < n) p[i] = 0.0f;
}

// ---------------------------------------------------------------------------
// QKV projection: O = X @ W + bias, via V_WMMA_F32_16X16X4_F32 (full fp32).
// grid = (N/16, 3), block = 256 (8 waves). blockIdx.y selects q/k/v.
// Also accumulates global sum-of-squares for q (ssq[0]) and k (ssq[1]).
__global__ __launch_bounds__(256) void qkv_wmma_kernel(
    const float* __restrict__ query, const float* __restrict__ source,
    const float* __restrict__ Wq, const float* __restrict__ bq,
    const float* __restrict__ Wk, const float* __restrict__ bk,
    const float* __restrict__ Wv, const float* __restrict__ bv,
    float* __restrict__ qo, float* __restrict__ ko, float* __restrict__ vo,
    float* __restrict__ ssq)
{
    const int mat = blockIdx.y;
    const float* X    = (mat == 0) ? query : source;
    const float* W    = (mat == 0) ? Wq : (mat == 1) ? Wk : Wv;
    const float* bias = (mat == 0) ? bq : (mat == 1) ? bk : bv;
    float*       O    = (mat == 0) ? qo : (mat == 1) ? ko : vo;

    const int tid     = threadIdx.x;
    const int rowBase = blockIdx.x * 16;

    __shared__ float As[16][IN_ + 4];   // +4 pad: A-frag reads hit distinct banks
    __shared__ float red[256];

    // stage 16x256 A tile (coalesced: thread t loads column t of each row)
#pragma unroll
    for (int i = 0; i < 16; ++i)
        As[i][tid] = X[(size_t)(rowBase + i) * IN_ + tid];
    __syncthreads();

    const int lane  = tid & 31;
    const int wv    = tid >> 5;        // wave id 0..7
    const int col16 = lane & 15;       // M index for A-frag, N index for B-frag
    const int hi    = lane >> 4;       // 0: K0/K1,  1: K2/K3
    const int kb    = hi * 2;
    const int cb0   = wv * 32;         // this wave's two 16-col output tiles
    const int col0  = cb0 + col16;
    const int col1  = cb0 + 16 + col16;

    v8f c0 = {};
    v8f c1 = {};

    for (int kt = 0; kt < IN_ / 4; ++kt) {
        const int kk = kt * 4 + kb;
        v2f a;
        a.x = As[col16][kt * 4 + kb];
        a.y = As[col16][kt * 4 + kb + 1];
        v2f b0, b1;
        b0.x = W[(size_t)kk * HD_ + col0];
        b0.y = W[(size_t)(kk + 1) * HD_ + col0];
        b1.x = W[(size_t)kk * HD_ + col1];
        b1.y = W[(size_t)(kk + 1) * HD_ + col1];
        c0 = __builtin_amdgcn_wmma_f32_16x16x4_f32(false, a, false, b0,
                                                   (short)0, c0, false, false);
        c1 = __builtin_amdgcn_wmma_f32_16x16x4_f32(false, a, false, b1,
                                                   (short)0, c1, false, false);
        if (kt + 2 < IN_ / 4)  // prefetch next weight lines (global_prefetch_b8)
            __builtin_prefetch(&W[(size_t)(kk + 8) * HD_ + col0], 0, 1);
    }

    const float bvv0 = bias[col0];
    const float bvv1 = bias[col1];
    float local = 0.0f;
#pragma unroll
    for (int i = 0; i < 8; ++i) {
        const int r = rowBase + i + 8 * hi;
        const float t0 = c0[i] + bvv0;
        const float t1 = c1[i] + bvv1;
        O[(size_t)r * HD_ + col0] = t0;
        O[(size_t)r * HD_ + col1] = t1;
        local += t0 * t0 + t1 * t1;
    }

    if (mat < 2) {  // global ||q||^2 / ||k||^2
        red[tid] = local;
        __syncthreads();
        for (int s = 128; s > 0; s >>= 1) {
            if (tid < s) red[tid] += red[tid + s];
            __syncthreads();
        }
        if (tid == 0) atomicAdd(&ssq[mat], red[0]);
    }
}

// ---------------------------------------------------------------------------
// degree histogram over col indices
__global__ void deg_kernel(const int* __restrict__ ei, float* __restrict__ deg) {
    int e = blockIdx.x * blockDim.x + threadIdx.x;
    if (e < E_) atomicAdd(&deg[ei[E_ + e]], 1.0f);
}

// ---------------------------------------------------------------------------
// vh[n,d] = sum_h v[n,h,d]   (head-summed values for the GCN scatter)
__global__ void vh_kernel(const float* __restrict__ v, float* __restrict__ vh) {
    int idx = blockIdx.x * blockDim.x + threadIdx.x;  // N*D threads
    int n = idx >> 6, d = idx & 63;
    const float* vp = v + (size_t)n * HD_ + d;
    vh[idx] = vp[0] + vp[D_] + vp[2 * D_] + vp[3 * D_];
}

// ---------------------------------------------------------------------------
// per-(b,h): kv[64][64] = k_norm^T v, ksum = sum_m k_norm, vsum = sum_m v
// one block per (b,h), 256 threads; each thread owns 16 kv entries.
__global__ __launch_bounds__(256) void kv_kernel(
    const float* __restrict__ k, const float* __restrict__ v,
    const float* __restrict__ ssq,
    float* __restrict__ kvo, float* __restrict__ ksum, float* __restrict__ vsum)
{
    const int bh = blockIdx.x;              // b*H + h
    const int b  = bh / H_;
    const int h  = bh % H_;
    const int tid = threadIdx.x;

    __shared__ float krow[D_];
    __shared__ float vrow[D_];

    const int kk  = tid >> 2;               // 0..63
    const int dvb = (tid & 3) * 16;         // 16-wide slice of d_v

    float acc[16];
#pragma unroll
    for (int j = 0; j < 16; ++j) acc[j] = 0.0f;
    float ks = 0.0f, vs = 0.0f;

    for (int m = 0; m < M_; ++m) {
        const size_t n = (size_t)b * M_ + m;
        if (tid < 64)       krow[tid]      = k[n * HD_ + h * D_ + tid];
        else if (tid < 128) vrow[tid - 64] = v[n * HD_ + h * D_ + (tid - 64)];
        __syncthreads();
        const float kv_ = krow[kk];
#pragma unroll
        for (int j = 0; j < 16; ++j) acc[j] += kv_ * vrow[dvb + j];
        if (tid < 64)       ks += krow[tid];
        else if (tid < 128) vs += vrow[tid - 64];
        __syncthreads();
    }

    const float inv_nk = rsqrtf(ssq[1]);    // fold k-normalization into kv/ksum
    float* kvp = kvo + (size_t)bh * D_ * D_;
#pragma unroll
    for (int j = 0; j < 16; ++j) kvp[kk * D_ + dvb + j] = acc[j] * inv_nk;
    if (tid < 64)       ksum[(size_t)bh * D_ + tid]        = ks * inv_nk;
    else if (tid < 128) vsum[(size_t)bh * D_ + (tid - 64)] = vs;
}

// ---------------------------------------------------------------------------
// attention: out[n,d] = (1/H) * sum_h (q_n·kv + vsum) / (q_n·ksum + n_nodes)
// one block per node, 64 threads (one per d).
__global__ __launch_bounds__(64) void attn_kernel(
    const float* __restrict__ q, const float* __restrict__ kv,
    const float* __restrict__ ksum, const float* __restrict__ vsum,
    const float* __restrict__ ssq, const int* __restrict__ n_nodes,
    float* __restrict__ out)
{
    const int n   = blockIdx.x;
    const int tid = threadIdx.x;
    const int b   = n / M_;
    const float inv_nq = rsqrtf(ssq[0]);
    const float nn = (float)n_nodes[b];

    __shared__ float qs[D_];
    float acc = 0.0f;

    for (int h = 0; h < H_; ++h) {
        const int bh = b * H_ + h;
        qs[tid] = q[(size_t)n * HD_ + h * D_ + tid] * inv_nq;
        __syncthreads();
        const float* kvp = kv   + (size_t)bh * D_ * D_;
        const float* ksp = ksum + (size_t)bh * D_;
        float num = vsum[(size_t)bh * D_ + tid];
        float den = nn;
#pragma unroll 8
        for (int kk = 0; kk < D_; ++kk) {
            const float qv = qs[kk];
            num += qv * kvp[kk * D_ + tid];
            den += qv * ksp[kk];
        }
        acc += num / den;
        __syncthreads();
    }
    out[(size_t)n * D_ + tid] = acc * (1.0f / H_);
}

// ---------------------------------------------------------------------------
// GCN scatter: out[col,d] += (w_e / sqrt(deg[col]*deg[row])) * (1/H) * vh[row,d]
// 64 threads per edge (2 waves), coalesced vh reads + float atomics into out.
__global__ __launch_bounds__(256) void gcn_kernel(
    const int* __restrict__ ei, const float* __restrict__ w,
    const float* __restrict__ deg, const float* __restrict__ vh,
    float* __restrict__ out)
{
    const int t = blockIdx.x * 256 + threadIdx.x;
    const int e = t >> 6;
    const int d = t & 63;
    if (e >= E_) return;
    const int r = ei[e];
    const int c = ei[E_ + e];
    const float dd  = deg[c] * deg[r];
    const float val = (dd > 0.0f) ? (w[e] * rsqrtf(dd)) : 0.0f;  // nan_to_num
    atomicAdd(&out[(size_t)c * D_ + d], val * (1.0f / H_) * vh[(size_t)r * D_ + d]);
}

// ---------------------------------------------------------------------------
extern "C" void kernel_launch(void* const* d_in, const int* in_sizes, int n_in,
                              void* d_out, int out_size, void* d_ws, size_t ws_size,
                              hipStream_t stream) {
    (void)in_sizes; (void)n_in; (void)out_size; (void)ws_size;

    const float* query  = (const float*)d_in[0];
    const float* source = (const float*)d_in[1];
    const float* ew     = (const float*)d_in[2];
    const float* Wq     = (const float*)d_in[3];
    const float* bq     = (const float*)d_in[4];
    const float* Wk     = (const float*)d_in[5];
    const float* bk     = (const float*)d_in[6];
    const float* Wv     = (const float*)d_in[7];
    const float* bv     = (const float*)d_in[8];
    const int*   nnod   = (const int*)d_in[9];
    const int*   ei     = (const int*)d_in[10];
    float* out = (float*)d_out;
    float* ws  = (float*)d_ws;

    float* q    = ws + O_Q;
    float* k    = ws + O_K;
    float* v    = ws + O_V;
    float* vh   = ws + O_VH;
    float* kv   = ws + O_KV;
    float* ksum = ws + O_KSUM;
    float* vsum = ws + O_VSUM;
    float* deg  = ws + O_DEG;
    float* ssq  = ws + O_SSQ;   // [0]=||q||^2, [1]=||k||^2

    // 1) zero deg + ssq (contiguous)
    zero_kernel<<<(N_ + 2 + 255) / 256, 256, 0, stream>>>(deg, N_ + 2);

    // 2) QKV projections via fp32 WMMA (+ global sumsq of q,k)
    qkv_wmma_kernel<<<dim3(N_ / 16, 3), 256, 0, stream>>>(
        query, source, Wq, bq, Wk, bk, Wv, bv, q, k, v, ssq);

    // 3) degree histogram
    deg_kernel<<<E_ / 256, 256, 0, stream>>>(ei, deg);

    // 4) head-summed values
    vh_kernel<<<(N_ * D_) / 256, 256, 0, stream>>>(v, vh);

    // 5) per-graph kv / ksum / vsum
    kv_kernel<<<B_ * H_, 256, 0, stream>>>(k, v, ssq, kv, ksum, vsum);

    // 6) attention -> writes out
    attn_kernel<<<N_, 64, 0, stream>>>(q, kv, ksum, vsum, ssq, nnod, out);

    // 7) GCN scatter -> atomicAdd into out
    gcn_kernel<<<(E_ * D_) / 256, 256, 0, stream>>>(ei, ew, deg, vh, out);
}